// FakeEdgePooling_53395033423882
// MI455X (gfx1250) — compile-verified
//
#include <hip/hip_runtime.h>
#include <hip/hip_bf16.h>

// Problem constants (from reference).
#define N_NODES   50000
#define N_EDGES   400000
#define C_DIM     128
#define K_DIM     256            // 2*C : concat(src,dst)
#define LN_EPS    1e-5f

typedef __attribute__((ext_vector_type(2))) float v2f;
typedef __attribute__((ext_vector_type(8))) float v8f;

// Block = 256 threads = 8 waves (wave32).  Each wave: 16 edges x 128 channels.
// Block tile = 128 edges.  Grid = E/128 = 3125 exactly.
//
// Dynamic LDS layout (floats):
//   [0      , 32768) : W1 staged as B-fragments.
//       ib(k,h,n,j) = k*512 + h*256 + n*2 + j   holds W1[n][4k+2h+j]
//       (ISA 7.12.2: f32 B 4x16 frag: VGPR j, lane-half h -> K = 4k + 2h + j)
//   [32768 , 65536) : gathered A panel, 4096 floats per wave.
//       ia(k,h,m,j) = k*64 + h*32 + m*2 + j     holds A[m][4k+2h+j]
//       where A[m][kk] = kk<128 ? x[src][kk] : x[dst][kk-128]
__global__ void __launch_bounds__(256)
edge_score_wmma_kernel(const float* __restrict__ x,
                       const int*   __restrict__ edge_index,   // [2*E]
                       const float* __restrict__ W1,           // [128 x 256] row-major
                       const float* __restrict__ b1,           // [128]
                       const float* __restrict__ ln_g,         // [128]
                       const float* __restrict__ ln_b,         // [128]
                       const float* __restrict__ W2,           // [128]
                       const float* __restrict__ b2,           // [1]
                       float*       __restrict__ logits)       // [E]
{
    extern __shared__ float smem[];

    const int tid  = threadIdx.x;
    const int wave = tid >> 5;       // 0..7
    const int lane = tid & 31;
    const int half = lane >> 4;      // lane-half (0: lanes 0-15, 1: lanes 16-31)
    const int ml   = lane & 15;

    const int e_base = blockIdx.x * 128 + wave * 16;   // first edge of this wave

    const int* __restrict__ srcIdx = edge_index;
    const int* __restrict__ dstIdx = edge_index + N_EDGES;

    // ---- Per-lane channel parameters for the 8 N-tiles this lane touches ----
    float gv[8], bbv[8], w2v[8], b1v[8];
#pragma unroll
    for (int nt = 0; nt < 8; ++nt) {
        const int ch = nt * 16 + ml;
        gv[nt]  = ln_g[ch];
        bbv[nt] = ln_b[ch];
        w2v[nt] = W2[ch];
        b1v[nt] = b1[ch];
    }
    const float b2v = b2[0];

    // ---- Stage W1 into LDS in B-fragment order (float4 loads, v2f stores) ---
    // unit u = (k,n): n = u&127 (low bits -> LDS banks spread), k = u>>7.
    for (int u = tid; u < 8192; u += 256) {
        const int n = u & 127;
        const int k = u >> 7;
        const float4 w = *(const float4*)(W1 + n * K_DIM + 4 * k);
        *(v2f*)(smem + k * 512 +       n * 2) = (v2f){w.x, w.y};   // h=0, j=0..1
        *(v2f*)(smem + k * 512 + 256 + n * 2) = (v2f){w.z, w.w};   // h=1, j=0..1
    }

    // ---- Stage this wave's gathered A panel (16 edges x 256) ----------------
    // Broadcast the wave's 16 src + 16 dst row ids through lane registers.
    const int rowreg = (half == 0) ? srcIdx[e_base + ml] : dstIdx[e_base + ml];

    float* sA = smem + 32768 + wave * 4096;
    // unit u4 = (k4,m): m = u4&15, k4 = u4>>4; one float4 = kk 4k4..4k4+3.
    for (int u4 = lane; u4 < 1024; u4 += 32) {
        const int m   = u4 & 15;
        const int k4  = u4 >> 4;
        const int kk0 = 4 * k4;
        const int row = __shfl(rowreg, m + ((kk0 >= 128) ? 16 : 0), 32);
        const float4 v = *(const float4*)(x + row * C_DIM + (kk0 & 127));
        *(v2f*)(sA + k4 * 64 +      m * 2) = (v2f){v.x, v.y};      // h=0, j=0..1
        *(v2f*)(sA + k4 * 64 + 32 + m * 2) = (v2f){v.z, v.w};      // h=1, j=0..1
    }

    __syncthreads();

    // ---- Main GEMM: 64 K-steps x 8 N-tiles of V_WMMA_F32_16X16X4_F32 --------
    v8f acc[8];
#pragma unroll
    for (int nt = 0; nt < 8; ++nt) acc[nt] = (v8f){0.f,0.f,0.f,0.f,0.f,0.f,0.f,0.f};

    const float* sAl = sA + half * 32 + ml * 2;             // per-lane A addr
    const float* sBl = smem + half * 256 + ml * 2;          // per-lane B addr

    for (int k = 0; k < 64; ++k) {
        const v2f a = *(const v2f*)(sAl + k * 64);
        v2f bf[8];
#pragma unroll
        for (int nt = 0; nt < 8; ++nt)
            bf[nt] = *(const v2f*)(sBl + k * 512 + nt * 32);
#pragma unroll
        for (int nt = 0; nt < 8; ++nt)
            acc[nt] = __builtin_amdgcn_wmma_f32_16x16x4_f32(
                false, a, false, bf[nt], (short)0, acc[nt], false, false);
    }

    // ---- Epilogue: +b1, ReLU, LayerNorm(128), dot(W2)+b2 --------------------
    // C/D layout: acc[nt][r] at lane L = h[row r + 8*(L>>4)][ch 16*nt + (L&15)]
#pragma unroll
    for (int r = 0; r < 8; ++r) {
        float hv[8];
        float sum = 0.f, sq = 0.f;
#pragma unroll
        for (int nt = 0; nt < 8; ++nt) {
            float v = acc[nt][r] + b1v[nt];
            v = v > 0.f ? v : 0.f;          // ReLU
            hv[nt] = v;
            sum += v;
            sq  += v * v;
        }
        // reduce across the 16 lanes of this half (masks < 16 stay in-half)
#pragma unroll
        for (int msk = 1; msk <= 8; msk <<= 1) {
            sum += __shfl_xor(sum, msk, 32);
            sq  += __shfl_xor(sq,  msk, 32);
        }
        const float mean = sum * (1.0f / 128.0f);
        const float var  = sq * (1.0f / 128.0f) - mean * mean;
        const float rstd = rsqrtf(var + LN_EPS);

        float dot = 0.f;
#pragma unroll
        for (int nt = 0; nt < 8; ++nt) {
            const float nrm = (hv[nt] - mean) * rstd * gv[nt] + bbv[nt];
            dot += nrm * w2v[nt];
        }
#pragma unroll
        for (int msk = 1; msk <= 8; msk <<= 1)
            dot += __shfl_xor(dot, msk, 32);

        if (ml == 0)
            logits[e_base + half * 8 + r] = dot + b2v;
    }
}

extern "C" void kernel_launch(void* const* d_in, const int* in_sizes, int n_in,
                              void* d_out, int out_size, void* d_ws, size_t ws_size,
                              hipStream_t stream) {
    (void)in_sizes; (void)n_in; (void)out_size; (void)d_ws; (void)ws_size;

    const float* x          = (const float*)d_in[0];
    const int*   edge_index = (const int*)  d_in[1];
    // d_in[2] = batch (unused by the reference computation)
    const float* W1         = (const float*)d_in[3];
    const float* b1         = (const float*)d_in[4];
    const float* ln_g       = (const float*)d_in[5];
    const float* ln_b       = (const float*)d_in[6];
    const float* W2         = (const float*)d_in[7];
    const float* b2         = (const float*)d_in[8];

    float* out = (float*)d_out;

    // Output tuple = (x, logits): pass x through with an async D2D copy.
    hipMemcpyAsync(out, x, (size_t)N_NODES * C_DIM * sizeof(float),
                   hipMemcpyDeviceToDevice, stream);

    constexpr int    blocks   = N_EDGES / 128;     // 3125, exact
    constexpr size_t lds_size = 65536 * sizeof(float);   // 256 KB of 320 KB/WGP

    edge_score_wmma_kernel<<<blocks, 256, lds_size, stream>>>(
        x, edge_index, W1, b1, ln_g, ln_b, W2, b2, out + (size_t)N_NODES * C_DIM);
}